// LocalEncoderNLP_37924561223763
// MI455X (gfx1250) — compile-verified
//
#include <hip/hip_runtime.h>
#include <hip/hip_bf16.h>

typedef __attribute__((ext_vector_type(2))) float v2f;
typedef __attribute__((ext_vector_type(4))) float v4f;
typedef __attribute__((ext_vector_type(8))) float v8f;
typedef __attribute__((ext_vector_type(4))) int   v4i;

#define B_BATCH 8
#define N_SEQ   16384
#define DIM     256
#define WIDTH   33
#define CENTER  16
#define KROWS   48          // 16 (left halo) + 16 (tile) + 16 (right halo) rows of x
#define XPITCH  264         // LDS row pitch in floats: 2*264 % 64 == 16 -> conflict-free B-frag reads
#define NBLK    (N_SEQ / 16)

#if __has_builtin(__builtin_amdgcn_global_load_async_to_lds_b128)
#define ASYNC_LDS 1
typedef __attribute__((address_space(1))) v4i* gas_v4i_ptr;   // global AS pointer to int4
typedef __attribute__((address_space(3))) v4i* lds_v4i_ptr;   // LDS AS pointer to int4
#endif

__global__ __launch_bounds__(256) void local_encoder_banded_wmma(
    const float* __restrict__ x,
    const float* __restrict__ size_in,
    const float* __restrict__ sm,
    float* __restrict__ out)
{
    __shared__ __align__(16) float X_lds[KROWS * XPITCH];   // 50688 B
    __shared__ float W_lds[16 * WIDTH];                     //  2112 B (16 rows x 33 taps)
    __shared__ float rcp_lds[16];

    const int tid = threadIdx.x;
    const int b   = blockIdx.x / NBLK;
    const int t   = (blockIdx.x % NBLK) * 16;   // first output row of this tile

    // ---- Stage x rows [t-16, t+32) into LDS; out-of-range sequence rows -> 0 ----
    const size_t xbase = (size_t)b * N_SEQ * DIM;
    #pragma unroll
    for (int it = 0; it < 12; ++it) {
        int idx = tid + it * 256;            // 0..3071  (48 rows * 64 float4)
        int r   = idx >> 6;                  // 0..47
        int c4  = (idx & 63) << 2;           // 0,4,...,252
        int g   = t - CENTER + r;            // global x row
        float* ldst = &X_lds[r * XPITCH + c4];
        if ((unsigned)g < (unsigned)N_SEQ) {
            const float* gsrc = x + xbase + (size_t)g * DIM + c4;
#ifdef ASYNC_LDS
            // CDNA5 async copy: global -> LDS without VGPR round-trip (ASYNCcnt)
            __builtin_amdgcn_global_load_async_to_lds_b128(
                (gas_v4i_ptr)gsrc, (lds_v4i_ptr)ldst, 0, 0);
#else
            *(v4f*)ldst = *(const v4f*)gsrc;
#endif
        } else {
            v4f z = {0.f, 0.f, 0.f, 0.f};
            *(v4f*)ldst = z;
        }
    }

    // ---- Stage the 16x33 weight block (contiguous in memory -> coalesced) ----
    const size_t smbase = ((size_t)b * N_SEQ + t) * WIDTH;
    #pragma unroll
    for (int idx = tid; idx < 16 * WIDTH; idx += 256)
        W_lds[idx] = sm[smbase + idx];

    // ---- Per-row reciprocal of denom ----
    if (tid < 16) {
        float s = size_in[(size_t)b * N_SEQ + t + tid];
        rcp_lds[tid] = 1.0f / fmaxf(s, 1e-6f);
    }

#ifdef ASYNC_LDS
#if __has_builtin(__builtin_amdgcn_s_wait_asynccnt)
    __builtin_amdgcn_s_wait_asynccnt(0);
#else
    asm volatile("s_wait_asynccnt 0x0" ::: "memory");
#endif
#endif
    __syncthreads();

    // ---- WMMA compute: each wave owns two 16-wide channel tiles ----
    const int lane = tid & 31;
    const int wave = tid >> 5;
    const int half = lane >> 4;      // 0: lanes 0-15, 1: lanes 16-31
    const int lm   = lane & 15;

    // A fragments from banded weights: lane holds A[lm][kk*4 + va + 2*half],
    // A[i][k] = W[i][k-i] when 0 <= k-i < 33, else 0  (ISA 32-bit 16x4 layout)
    v2f afrag[12];
    #pragma unroll
    for (int kk = 0; kk < 12; ++kk) {
        int k0 = kk * 4 + 2 * half;
        #pragma unroll
        for (int e = 0; e < 2; ++e) {
            int  off = k0 + e - lm;
            bool vld = (unsigned)off < (unsigned)WIDTH;
            float w  = W_lds[lm * WIDTH + (vld ? off : 0)];
            if (e == 0) afrag[kk].x = vld ? w : 0.f;
            else        afrag[kk].y = vld ? w : 0.f;
        }
    }

    const int c0 = wave * 32;        // channel tiles [c0, c0+16) and [c0+16, c0+32)
    v8f acc0 = {};
    v8f acc1 = {};
    #pragma unroll
    for (int kk = 0; kk < 12; ++kk) {
        int kb = kk * 4 + 2 * half;
        // B fragments: lane holds X[kb + vb][col + lm]  (ISA 32-bit 4x16 layout)
        v2f b0, b1;
        b0.x = X_lds[(kb + 0) * XPITCH + c0 + lm];
        b0.y = X_lds[(kb + 1) * XPITCH + c0 + lm];
        b1.x = X_lds[(kb + 0) * XPITCH + c0 + 16 + lm];
        b1.y = X_lds[(kb + 1) * XPITCH + c0 + 16 + lm];
        // D = A(16x4) * B(4x16) + C ; two independent acc chains hide WMMA RAW latency
        acc0 = __builtin_amdgcn_wmma_f32_16x16x4_f32(
                   false, afrag[kk], false, b0, (short)0, acc0, false, false);
        acc1 = __builtin_amdgcn_wmma_f32_16x16x4_f32(
                   false, afrag[kk], false, b1, (short)0, acc1, false, false);
    }

    // ---- Scale by 1/denom and store (C/D layout: VGPR v -> row v + 8*half, col lm) ----
    const size_t obase = ((size_t)b * N_SEQ + t) * DIM;
    #pragma unroll
    for (int v = 0; v < 8; ++v) {
        int   m = v + 8 * half;
        float r = rcp_lds[m];
        out[obase + (size_t)m * DIM + c0 + lm]      = acc0[v] * r;
        out[obase + (size_t)m * DIM + c0 + 16 + lm] = acc1[v] * r;
    }
}

extern "C" void kernel_launch(void* const* d_in, const int* in_sizes, int n_in,
                              void* d_out, int out_size, void* d_ws, size_t ws_size,
                              hipStream_t stream) {
    const float* x    = (const float*)d_in[0];   // (B, N, D) f32
    const float* sz   = (const float*)d_in[1];   // (B, N, 1) f32
    const float* sm   = (const float*)d_in[2];   // (B, N, 33) f32
    float*       out  = (float*)d_out;           // (B, N, D) f32

    dim3 grid(B_BATCH * NBLK);   // 8192 workgroups
    dim3 block(256);             // 8 waves (wave32)
    local_encoder_banded_wmma<<<grid, block, 0, stream>>>(x, sz, sm, out);
}